// VQC_Layer_43817256354119
// MI455X (gfx1250) — compile-verified
//
#include <hip/hip_runtime.h>

typedef __attribute__((ext_vector_type(2))) float v2f;
typedef __attribute__((ext_vector_type(8))) float v8f;
typedef __attribute__((ext_vector_type(4))) int   v4i;

#define DIM 256
#define NQ  8
#define LSTR 260   // padded LDS row stride (floats): 4*(L%16) bank skew -> conflict-free

#define AS1 __attribute__((address_space(1)))
#define AS3 __attribute__((address_space(3)))

#if defined(__has_builtin)
#  if __has_builtin(__builtin_amdgcn_global_load_async_to_lds_b128)
#    define HAVE_ASYNC_LDS 1
#  else
#    define HAVE_ASYNC_LDS 0
#  endif
#else
#  define HAVE_ASYNC_LDS 0
#endif

#if HAVE_ASYNC_LDS
#  if __has_builtin(__builtin_amdgcn_s_wait_asynccnt)
#    define WAIT_ASYNC(n) __builtin_amdgcn_s_wait_asynccnt(n)
#  else
#    define WAIT_ASYNC(n) asm volatile("s_wait_asynccnt " #n ::: "memory")
#  endif
#else
#  define WAIT_ASYNC(n) ((void)0)
#endif

// ---------------------------------------------------------------------------
// Kernel A: build the full-circuit unitary U (256x256 complex, row-major,
// split re/im planes) once per launch. Single workgroup, 256 threads.
// ---------------------------------------------------------------------------
__global__ void build_unitary(const float* __restrict__ params,
                              float* __restrict__ Ur,
                              float* __restrict__ Ui) {
  const int tid = threadIdx.x;

  for (int e = tid; e < DIM * DIM; e += 256) {
    int r = e >> 8, c = e & 255;
    Ur[e] = (r == c) ? 1.f : 0.f;
    Ui[e] = 0.f;
  }
  __threadfence();
  __syncthreads();

  for (int l = 0; l < 2; ++l) {
    for (int q = 0; q < NQ; ++q) {
      float th = 0.5f * params[(l * NQ + q) * 2 + 0];   // RY angle / 2
      float ph = 0.5f * params[(l * NQ + q) * 2 + 1];   // RZ angle / 2
      float sy, cy, sp, cp;
      __sincosf(th, &sy, &cy);
      __sincosf(ph, &sp, &cp);
      const float u00r =  cp * cy, u00i = -sp * cy;
      const float u01r = -cp * sy, u01i =  sp * sy;
      const float u10r =  cp * sy, u10i =  sp * sy;
      const float u11r =  cp * cy, u11i =  sp * cy;

      const int bitpos = 7 - q;          // qubit 0 is MSB
      const int mask   = 1 << bitpos;

      for (int idx = tid; idx < 128 * 256; idx += 256) {
        int pairI = idx >> 8;
        int col   = idx & 255;
        int low   = pairI & (mask - 1);
        int high  = (pairI >> bitpos) << (bitpos + 1);
        int r0    = high | low;
        int r1    = r0 | mask;
        int e0 = (r0 << 8) | col, e1 = (r1 << 8) | col;
        float ar = Ur[e0], ai = Ui[e0];
        float br = Ur[e1], bi = Ui[e1];
        float nar = u00r*ar - u00i*ai + u01r*br - u01i*bi;
        float nai = u00r*ai + u00i*ar + u01r*bi + u01i*br;
        float nbr = u10r*ar - u10i*ai + u11r*br - u11i*bi;
        float nbi = u10r*ai + u10i*ar + u11r*bi + u11i*br;
        Ur[e0] = nar; Ui[e0] = nai;
        Ur[e1] = nbr; Ui[e1] = nbi;
      }
      __threadfence();
      __syncthreads();
    }
    for (int e = tid; e < DIM * DIM; e += 256) {
      int r = e >> 8;
      int p = (__popc(r & (r >> 1)) + ((r & (r >> 7)) & 1)) & 1;
      if (p) { Ur[e] = -Ur[e]; Ui[e] = -Ui[e]; }
    }
    __threadfence();
    __syncthreads();
  }
}

// ---------------------------------------------------------------------------
// Kernel B: 4 waves per workgroup, 64 batch rows per workgroup.
//   - psi block (64 rows, padded stride) built in LDS
//   - B tile (16 U-rows, re/im) async-copied to LDS, double-buffered; shared
//     by all 4 waves -> 4x less L2 traffic than per-wave streaming
//   - complex GEMM via V_WMMA_F32_16X16X4_F32, fused |.|^2 + sign readout
//   - LDS row stride 260 floats: fragment loads spread lanes over all 64
//     banks (4*(lane%16)+2*(lane/16) skew) -> conflict-free ds_load_b64
// Dynamic LDS: psi 2*64*260*4 + B 2*2*16*260*4 + cs 4 KB ~= 204 KB (<320 KB)
// ---------------------------------------------------------------------------
__global__ void vqc_gemm_readout(const float* __restrict__ inputs,
                                 const float* __restrict__ Ur,
                                 const float* __restrict__ Ui,
                                 float* __restrict__ out) {
  extern __shared__ float smem[];
  float* psiR = smem;                        // [64][LSTR]
  float* psiI = psiR + 64 * LSTR;            // [64][LSTR]
  float* BR   = psiI + 64 * LSTR;            // [2][16*LSTR]
  float* BI   = BR + 2 * 16 * LSTR;          // [2][16*LSTR]
  float* sCv  = BI + 2 * 16 * LSTR;          // [64][8]
  float* sSv  = sCv + 64 * NQ;               // [64][8]

  const int tid   = threadIdx.x;             // 0..127
  const int lane  = tid & 31;
  const int wave  = tid >> 5;
  const int wgrow = blockIdx.x * 64;

  // --- stage 1: cos/sin of half-angles for 64 rows ---
  for (int idx = tid; idx < 64 * NQ; idx += 128) {
    int row = idx >> 3, q = idx & 7;
    float s, c;
    __sincosf(0.5f * inputs[(wgrow + row) * NQ + q], &s, &c);
    sCv[idx] = c; sSv[idx] = s;
  }
  __syncthreads();

  // --- stage 2: product-state amplitudes, phase = (-i)^popcount ---
  for (int e = tid; e < 64 * DIM; e += 128) {
    int row = e >> 8, i = e & 255;
    float m = 1.f;
#pragma unroll
    for (int q = 0; q < NQ; ++q) {
      m *= ((i >> (7 - q)) & 1) ? sSv[row * NQ + q] : sCv[row * NQ + q];
    }
    int pc = __popc(i) & 3;
    float re = 0.f, im = 0.f;
    if      (pc == 0) re =  m;
    else if (pc == 1) im = -m;
    else if (pc == 2) re = -m;
    else              im =  m;
    psiR[row * LSTR + i] = re;
    psiI[row * LSTR + i] = im;
  }
  __syncthreads();

  // --- B tile staging: 16 contiguous U rows (16 KB/plane) -> padded LDS ---
  auto issue_tile = [&](int nt, int buf) {
    const float* gR = Ur + nt * 16 * DIM;
    const float* gI = Ui + nt * 16 * DIM;
    float* lR = BR + buf * 16 * LSTR;
    float* lI = BI + buf * 16 * LSTR;
    // 16 rows x 64 b128-chunks per plane; dst 1040-B row stride (16-B aligned)
#if HAVE_ASYNC_LDS
    for (int t = tid; t < 16 * 64; t += 128) {
      int row = t >> 6, c4 = (t & 63) * 4;
      __builtin_amdgcn_global_load_async_to_lds_b128(
          (AS1 v4i*)(gR + row * DIM + c4), (AS3 v4i*)(lR + row * LSTR + c4), 0, 0);
      __builtin_amdgcn_global_load_async_to_lds_b128(
          (AS1 v4i*)(gI + row * DIM + c4), (AS3 v4i*)(lI + row * LSTR + c4), 0, 0);
    }
#else
    for (int t = tid; t < 16 * 64; t += 128) {
      int row = t >> 6, c4 = (t & 63) * 4;
      *(float4*)(lR + row * LSTR + c4) = *(const float4*)(gR + row * DIM + c4);
      *(float4*)(lI + row * LSTR + c4) = *(const float4*)(gI + row * DIM + c4);
    }
#endif
  };

  // --- stage 3: WMMA GEMM + fused readout, double-buffered B ---
  const int mrow = lane & 15;   // A: M index / B: N index
  const int hi   = lane >> 4;   // K half selector (K = 2*hi + v)

  float PA[8], PB[8];
#pragma unroll
  for (int v = 0; v < 8; ++v) { PA[v] = 0.f; PB[v] = 0.f; }

  const float* aR = psiR + (wave * 16 + mrow) * LSTR;
  const float* aI = psiI + (wave * 16 + mrow) * LSTR;

  issue_tile(0, 0);
  for (int nt = 0; nt < 16; ++nt) {
    if (nt < 15) {
      issue_tile(nt + 1, (nt + 1) & 1);
      WAIT_ASYNC(16);          // leave only next-stage ops outstanding
    } else {
      WAIT_ASYNC(0);
    }
    __syncthreads();           // all waves' portions of current tile visible

    const float* bR = BR + (nt & 1) * 16 * LSTR + mrow * LSTR;
    const float* bI = BI + (nt & 1) * 16 * LSTR + mrow * LSTR;

    v8f acc_rr = {}, acc_ii = {}, acc_ri = {}, acc_ir = {};
    for (int kb = 0; kb < DIM; kb += 4) {
      const int ko = kb + 2 * hi;
      v2f ar = *(const v2f*)(aR + ko);
      v2f ai = *(const v2f*)(aI + ko);
      v2f br = *(const v2f*)(bR + ko);
      v2f bi = *(const v2f*)(bI + ko);
      acc_rr = __builtin_amdgcn_wmma_f32_16x16x4_f32(false, ar, false, br,
                                                     (short)0, acc_rr, false, false);
      acc_ii = __builtin_amdgcn_wmma_f32_16x16x4_f32(false, ai, false, bi,
                                                     (short)0, acc_ii, false, false);
      acc_ri = __builtin_amdgcn_wmma_f32_16x16x4_f32(false, ar, false, bi,
                                                     (short)0, acc_ri, false, false);
      acc_ir = __builtin_amdgcn_wmma_f32_16x16x4_f32(false, ai, false, br,
                                                     (short)0, acc_ir, false, false);
    }

    const int j = nt * 16 + mrow;                                // state index
    const float sa = (((j >> 7) ^ (j >> 6)) & 1) ? -1.f : 1.f;   // Z0*Z1
    const float sb = (__popc(j & 63) & 1)        ? -1.f : 1.f;   // Z2..Z7

#pragma unroll
    for (int v = 0; v < 8; ++v) {
      float re = acc_rr[v] - acc_ii[v];
      float im = acc_ri[v] + acc_ir[v];
      float pr = re * re + im * im;
      PA[v] += pr * sa;
      PB[v] += pr * sb;
    }
    __syncthreads();           // done reading tile before it is overwritten
  }

  // reduce across the 16 columns held by each 16-lane half
#pragma unroll
  for (int v = 0; v < 8; ++v) {
#pragma unroll
    for (int m = 1; m <= 8; m <<= 1) {
      PA[v] += __shfl_xor(PA[v], m, 32);
      PB[v] += __shfl_xor(PB[v], m, 32);
    }
  }

  if ((lane & 15) == 0) {
    const int roff = hi * 8;   // lane 0 -> rows 0..7, lane 16 -> rows 8..15
    const int base = wgrow + wave * 16 + roff;
#pragma unroll
    for (int v = 0; v < 8; ++v) {
      out[(base + v) * 2 + 0] = PA[v];
      out[(base + v) * 2 + 1] = PB[v];
    }
  }
}

// ---------------------------------------------------------------------------
extern "C" void kernel_launch(void* const* d_in, const int* in_sizes, int n_in,
                              void* d_out, int out_size, void* d_ws, size_t ws_size,
                              hipStream_t stream) {
  const float* inputs = (const float*)d_in[0];   // (16384, 8) f32
  const float* params = (const float*)d_in[1];   // (1, 32)    f32
  float* Ur = (float*)d_ws;                      // 256*256 f32
  float* Ui = Ur + DIM * DIM;                    // 256*256 f32 (512 KB total)
  float* out = (float*)d_out;                    // (16384, 2) f32

  hipLaunchKernelGGL(build_unitary, dim3(1), dim3(256), 0, stream, params, Ur, Ui);

  const int batch = in_sizes[0] / NQ;            // 16384
  const size_t ldsBytes =
      (size_t)(2 * 64 * LSTR + 4 * 16 * LSTR + 2 * 64 * NQ) * sizeof(float); // ~204 KB
  hipLaunchKernelGGL(vqc_gemm_readout, dim3(batch / 64), dim3(128), ldsBytes,
                     stream, inputs, Ur, Ui, out);
}